// AURCLoss_25125558681838
// MI455X (gfx1250) — compile-verified
//
#include <hip/hip_runtime.h>
#include <hip/hip_bf16.h>

// ---------------------------------------------------------------------------
// AURC loss for B=8192, C=32000 f32 logits (wave32 / gfx1250).
//   score_i = 1/S_i,  S_i = sum_j exp(x_ij - max_i)
//   loss_i  = max_i + log(S_i) - x_i[target_i]
//   out     = (1/B) * sum_{k=0}^{B-2} csum(sorted_losses)[k] / (k+1)
// Phase 1 is HBM-bandwidth bound (~1.05 GB -> ~45us @ 23.3 TB/s).
// ---------------------------------------------------------------------------

typedef __attribute__((ext_vector_type(2))) float v2f;
typedef __attribute__((ext_vector_type(8))) float v8f;

// Sum 32 lane values across a wave. Uses V_WMMA_F32_16X16X4_F32 with a
// ones B-matrix: A[m,k] = {p_m, 0, p_{m+16}, 0} -> D[m,n] = p_m + p_{m+16}.
// Summing the 8 D-VGPRs in a lane gives half-sums; one xor-16 shuffle
// completes the reduction on every lane. Requires EXEC == all ones.
__device__ inline float wave_sum32(float x) {
#if __has_builtin(__builtin_amdgcn_wmma_f32_16x16x4_f32)
  v2f a; a[0] = x;    a[1] = 0.0f;
  v2f b; b[0] = 1.0f; b[1] = 1.0f;
  v8f c = {};
  c = __builtin_amdgcn_wmma_f32_16x16x4_f32(
      /*neg_a=*/false, a, /*neg_b=*/false, b,
      /*c_mod=*/(short)0, c, /*reuse_a=*/false, /*reuse_b=*/false);
  float half = c[0] + c[1] + c[2] + c[3] + c[4] + c[5] + c[6] + c[7];
  return half + __shfl_xor(half, 16, 32);
#else
  for (int off = 16; off > 0; off >>= 1) x += __shfl_xor(x, off, 32);
  return x;
#endif
}

// ---------------------------------------------------------------------------
// Kernel 1: one block per row; 256 threads (8 wave32) stream the row with
// b128 loads and a single-pass online logsumexp.
// ---------------------------------------------------------------------------
__global__ __launch_bounds__(256) void aurc_row_stats(
    const float* __restrict__ x, const long long* __restrict__ target,
    float* __restrict__ scores, float* __restrict__ losses, int C) {
  const int row = blockIdx.x;
  const int tid = threadIdx.x;
  const float* xrow = x + (size_t)row * (size_t)C;

  __shared__ float s_xt;
  __shared__ float wm[8], ws[8];
  if (tid == 0) {
    long long t = target[row];
    s_xt = xrow[t];
  }

  // Online (max, sum-exp) over strided float4 elements: 1 exp per element.
  const float4* xv = (const float4*)xrow;
  const int n4 = C >> 2;  // 8000
  float m = -3.402823466e38f;
  float s = 0.0f;
  for (int i = tid; i < n4; i += 256) {
    __builtin_prefetch(xv + i + 2048, 0, 0);  // global_prefetch_b8
    float4 v = xv[i];
    float e[4] = {v.x, v.y, v.z, v.w};
#pragma unroll
    for (int q = 0; q < 4; ++q) {
      float vq = e[q];
      if (vq <= m) {
        s += __expf(vq - m);
      } else {
        s = s * __expf(m - vq) + 1.0f;  // exp(-huge) underflows to 0 on first hit
        m = vq;
      }
    }
  }

  // wave32 pairwise (m,s) combine
  for (int off = 16; off > 0; off >>= 1) {
    float m2 = __shfl_xor(m, off, 32);
    float s2 = __shfl_xor(s, off, 32);
    float nm = fmaxf(m, m2);
    s = s * __expf(m - nm) + s2 * __expf(m2 - nm);
    m = nm;
  }
  const int wave = tid >> 5;
  if ((tid & 31) == 0) { wm[wave] = m; ws[wave] = s; }
  __syncthreads();

  if (tid == 0) {
    float M = wm[0], S = ws[0];
#pragma unroll
    for (int w = 1; w < 8; ++w) {
      float nm = fmaxf(M, wm[w]);
      S = S * __expf(M - nm) + ws[w] * __expf(wm[w] - nm);
      M = nm;
    }
    scores[row] = 1.0f / S;                  // exp(max log_softmax)
    losses[row] = M + __logf(S) - s_xt;      // -log_softmax[target]
  }
}

// ---------------------------------------------------------------------------
// Kernel 2: single 1024-thread block. Bitonic-sort 8192 (score, loss) pairs
// in LDS (descending by score), scan losses, accumulate csum[k]/(k+1),
// reduce with WMMA wave sums.
// ---------------------------------------------------------------------------
#define NSORT 8192
#define NTHR  1024

__global__ __launch_bounds__(NTHR) void aurc_finalize(
    const float* __restrict__ scores, const float* __restrict__ losses,
    float* __restrict__ out) {
  __shared__ float sc[NSORT];  // scores during sort; reused as scan scratch after
  __shared__ float ls[NSORT];
  const int tid = threadIdx.x;

  for (int i = tid; i < NSORT; i += NTHR) { sc[i] = scores[i]; ls[i] = losses[i]; }
  __syncthreads();

  // Bitonic sort, descending by sc[]; ls[] carried along.
  for (int k = 2; k <= NSORT; k <<= 1) {
    for (int j = k >> 1; j > 0; j >>= 1) {
      for (int i = tid; i < NSORT; i += NTHR) {
        int ixj = i ^ j;
        if (ixj > i) {
          float a = sc[i], b = sc[ixj];
          bool desc = ((i & k) == 0);
          if (desc ? (a < b) : (a > b)) {
            sc[i] = b; sc[ixj] = a;
            float la = ls[i], lb = ls[ixj];
            ls[i] = lb; ls[ixj] = la;
          }
        }
      }
      __syncthreads();
    }
  }

  // Each thread owns 8 contiguous sorted losses.
  const int PER = NSORT / NTHR;  // 8
  float loc[PER];
  float tsumv = 0.0f;
#pragma unroll
  for (int e = 0; e < PER; ++e) { loc[e] = ls[tid * PER + e]; tsumv += loc[e]; }

  // Inclusive Hillis-Steele scan of the 1024 per-thread totals (reuse sc[]).
  float* tsum = sc;
  tsum[tid] = tsumv;
  __syncthreads();
  for (int off = 1; off < NTHR; off <<= 1) {
    float v = (tid >= off) ? tsum[tid - off] : 0.0f;
    __syncthreads();
    tsum[tid] += v;
    __syncthreads();
  }
  float run = tsum[tid] - tsumv;  // exclusive prefix

  float contrib = 0.0f;
#pragma unroll
  for (int e = 0; e < PER; ++e) {
    int k = tid * PER + e;
    run += loc[e];
    if (k < NSORT - 1) contrib += run / (float)(k + 1);
  }

  // Two-level reduction of 1024 contributions: WMMA wave sums + LDS.
  float wsum = wave_sum32(contrib);
  float* wred = sc + 2048;  // disjoint from tsum[0..1023]
  const int wave = tid >> 5, lane = tid & 31;
  if (lane == 0) wred[wave] = wsum;
  __syncthreads();
  if (wave == 0) {
    float total = wave_sum32(wred[lane]);  // 32 wave-partials in one wave
    if (lane == 0) out[0] = total / (float)NSORT;
  }
}

// ---------------------------------------------------------------------------
extern "C" void kernel_launch(void* const* d_in, const int* in_sizes, int n_in,
                              void* d_out, int out_size, void* d_ws, size_t ws_size,
                              hipStream_t stream) {
  const float* input = (const float*)d_in[0];
  const long long* target = (const long long*)d_in[1];  // int64 targets
  float* out = (float*)d_out;

  const int B = in_sizes[1];           // 8192
  const int C = in_sizes[0] / B;       // 32000

  float* scores = (float*)d_ws;
  float* losses = scores + B;

  aurc_row_stats<<<B, 256, 0, stream>>>(input, target, scores, losses, C);
  aurc_finalize<<<1, NTHR, 0, stream>>>(scores, losses, out);
}